// SpatialThresholdSelector_89103391523407
// MI455X (gfx1250) — compile-verified
//
#include <hip/hip_runtime.h>
#include <cmath>

// ---------------- problem constants ----------------
#define BATCH   64
#define NPATCH  784
#define NPAD    1024
#define KSEL    392          // int(784 * 0.5)
#define SIDE    28
#define HW      448
#define PIX     (HW*HW)      // 200704 floats per image
#define F4_IMG  (PIX/4)      // 50176 float4 per image
#define PB      28           // partial blocks per batch (50176/28 = 1792 f4, /256 = 7 chunks)
#define F4_BLK  (F4_IMG/PB)  // 1792
#define CHUNK   256
#define NCHUNK  (F4_BLK/CHUNK) // 7
#define DMODEL  768
#define THRESH  0.3f

typedef __attribute__((ext_vector_type(2))) float v2f;
typedef __attribute__((ext_vector_type(8))) float v8f;
typedef __attribute__((ext_vector_type(4))) int   v4i;

typedef __attribute__((address_space(1))) v4i* gv4_t;   // global int4*
typedef __attribute__((address_space(3))) v4i* lv4_t;   // LDS int4*

// =====================================================================
// Kernel A: per-batch partial COG sums over line_drawing.
// Streams pixels global -> LDS with async-to-LDS (double buffered, each
// thread reads back only the LDS bytes it requested -> no barriers,
// per-wave sync via s_wait_asynccnt only).  Writes (sum, sum*y, sum*x)
// per block to workspace.
// =====================================================================
__device__ __forceinline__ void issue_async(const float4* g, float4* l) {
    v4i* gp = (v4i*)(void*)const_cast<float4*>(g);
    v4i* lp = (v4i*)(void*)l;
    __builtin_amdgcn_global_load_async_to_lds_b128((gv4_t)gp, (lv4_t)lp, 0, 0);
}

__global__ __launch_bounds__(256) void cog_partial_kernel(
        const float* __restrict__ line, float* __restrict__ partials) {
    __shared__ float4 buf[2][CHUNK];
    __shared__ float  red[3][256];

    const int tid = threadIdx.x;
    const int b   = blockIdx.x / PB;
    const int bi  = blockIdx.x % PB;
    const float4* img  = (const float4*)(line + (size_t)b * PIX);
    const int     base = bi * F4_BLK;
    const float   inv  = 1.0f / 447.0f;   // linspace(0,1,448) step

    float s = 0.0f, sy = 0.0f, sx = 0.0f;

    issue_async(img + base + 0 * CHUNK + tid, &buf[0][tid]);
    for (int c = 0; c < NCHUNK; ++c) {
        if (c + 1 < NCHUNK) {
            issue_async(img + base + (c + 1) * CHUNK + tid, &buf[(c + 1) & 1][tid]);
            asm volatile("s_wait_asynccnt 0x1" ::: "memory");
        } else {
            asm volatile("s_wait_asynccnt 0x0" ::: "memory");
        }
        float4 v = buf[c & 1][tid];
        int q  = base + c * CHUNK + tid;   // float4 index within image
        int h  = q / (HW / 4);             // 112 float4 per row (448 % 4 == 0)
        int w0 = (q % (HW / 4)) * 4;
        float y    = (float)h * inv;
        float lsum = v.x + v.y + v.z + v.w;
        float xsum = v.x * ((float)(w0    ) * inv)
                   + v.y * ((float)(w0 + 1) * inv)
                   + v.z * ((float)(w0 + 2) * inv)
                   + v.w * ((float)(w0 + 3) * inv);
        s  += lsum;
        sy += y * lsum;
        sx += xsum;
    }

    // deterministic block tree-reduction (no FP atomics anywhere)
    red[0][tid] = s; red[1][tid] = sy; red[2][tid] = sx;
    __syncthreads();
    for (int off = 128; off > 0; off >>= 1) {
        if (tid < off) {
            red[0][tid] += red[0][tid + off];
            red[1][tid] += red[1][tid + off];
            red[2][tid] += red[2][tid + off];
        }
        __syncthreads();
    }
    if (tid == 0) {
        float* o = partials + (size_t)blockIdx.x * 3;
        o[0] = red[0][0]; o[1] = red[1][0]; o[2] = red[2][0];
    }
}

// =====================================================================
// Kernel B: one block per batch.
//  1) WMMA (f32 16x16x4, A = ones) reduces the 28x3 partial matrix to
//     column sums: C[0,n] = sum_k B[k,n]  -> (total, sum*y, sum*x).
//  2) gaussian-weighted scores into LDS.
//  3) one bitonic sort of 1024 (val,idx) pairs with a comparator that
//     reproduces the reference's combined selection; first 392 = sel.
// =====================================================================
__device__ __forceinline__ bool before_cmp(float va, int ia, float vb, int ib,
                                           bool split) {
    if (split) {
        bool aa = va > THRESH, ab = vb > THRESH;
        if (aa != ab) return aa;       // above-threshold entries first
        if (aa) return ia < ib;        // among above: ascending index
    }
    if (va > vb) return true;          // descending value
    if (va < vb) return false;
    return ia < ib;                    // stable: lower index first (top_k tie rule)
}

__global__ __launch_bounds__(512) void select_kernel(
        const float* __restrict__ partials, const float* __restrict__ scores,
        int* __restrict__ sel) {
    __shared__ float ldsVal[NPAD];
    __shared__ int   ldsIdx[NPAD];
    __shared__ float cogLDS[3];
    __shared__ int   countSh;

    const int tid = threadIdx.x;
    const int b   = blockIdx.x;
    if (tid == 0) countSh = 0;

    // ---- matrix-engine column-sum of partials[28][3] ----
    if (tid < 32) {                       // wave 0, EXEC all-ones inside
        v8f acc = {};
        v2f a;  a[0] = 1.0f; a[1] = 1.0f; // A(16x4) = all ones
        const int n  = tid & 15;          // B column
        const int kb = (tid >> 4) * 2;    // K rows held by this half-wave
        const float* pbase = partials + (size_t)b * PB * 3;
        for (int it = 0; it < PB / 4; ++it) {   // 7 accumulating WMMAs
            int p0 = it * 4 + kb;
            v2f bm;
            bm[0] = (n < 3) ? pbase[(p0    ) * 3 + n] : 0.0f;
            bm[1] = (n < 3) ? pbase[(p0 + 1) * 3 + n] : 0.0f;
            acc = __builtin_amdgcn_wmma_f32_16x16x4_f32(
                      false, a, false, bm, (short)0, acc, false, false);
        }
        // C row M=0 lives in acc[0] on lanes 0..15 (N = lane)
        if (tid < 3) cogLDS[tid] = acc[0];
    }
    __syncthreads();

    const float total = cogLDS[0] + 1e-6f;
    const float cy = cogLDS[1] / total;
    const float cx = cogLDS[2] / total;
    const float inv27 = 1.0f / 27.0f;

    for (int i = tid; i < NPAD; i += 512) {
        if (i < NPATCH) {
            float gy = (float)(i / SIDE) * inv27;
            float gx = (float)(i % SIDE) * inv27;
            float dy = gy - cy, dx = gx - cx;
            float w  = expf(-8.0f * (dy * dy + dx * dx)); // 1/(2*0.25^2) = 8
            float wt = scores[(size_t)b * NPATCH + i] * w;
            ldsVal[i] = wt; ldsIdx[i] = i;
            if (wt > THRESH) atomicAdd(&countSh, 1);      // integer LDS atomic
        } else {
            ldsVal[i] = -__builtin_inff(); ldsIdx[i] = i; // pads sort last
        }
    }
    __syncthreads();

    const int  count = countSh;
    const bool split = (count > 0) && (count < KSEL);

    // bitonic sort, 1024 elements, 512 threads
    for (int size = 2; size <= NPAD; size <<= 1) {
        for (int stride = size >> 1; stride > 0; stride >>= 1) {
            __syncthreads();
            #pragma unroll
            for (int half = 0; half < 2; ++half) {
                int e = tid + half * 512;
                int p = e ^ stride;
                if (p > e) {
                    float ve = ldsVal[e], vp = ldsVal[p];
                    int   ie = ldsIdx[e], ip = ldsIdx[p];
                    bool up     = ((e & size) == 0);
                    bool pFirst = before_cmp(vp, ip, ve, ie, split);
                    if (pFirst == up) {
                        ldsVal[e] = vp; ldsVal[p] = ve;
                        ldsIdx[e] = ip; ldsIdx[p] = ie;
                    }
                }
            }
        }
    }
    __syncthreads();
    if (tid < KSEL) sel[(size_t)b * KSEL + tid] = ldsIdx[tid];
}

// =====================================================================
// Kernel C: gather + add, one block per output row, float4 everywhere.
//   out[b,j,:] = magno[b, sel[b,j], :] + pos[0, sel[b,j]+1, :]
// =====================================================================
__global__ __launch_bounds__(192) void gather_kernel(
        const float* __restrict__ magno, const float* __restrict__ pos,
        const int* __restrict__ sel, float* __restrict__ out) {
    const int row = blockIdx.x;          // b*KSEL + j
    const int b   = row / KSEL;
    const int s   = sel[row];
    const int t   = threadIdx.x;         // 192 threads * float4 = 768 floats

    const float4* src = (const float4*)(magno + ((size_t)b * NPATCH + s) * DMODEL);
    const float4* pe  = (const float4*)(pos   + ((size_t)(s + 1)) * DMODEL);
    float4*       dst = (float4*)(out + (size_t)row * DMODEL);

    float4 a = src[t], q = pe[t];
    dst[t] = make_float4(a.x + q.x, a.y + q.y, a.z + q.z, a.w + q.w);
}

// =====================================================================
extern "C" void kernel_launch(void* const* d_in, const int* in_sizes, int n_in,
                              void* d_out, int out_size, void* d_ws, size_t ws_size,
                              hipStream_t stream) {
    const float* magno  = (const float*)d_in[0]; // [64, 784, 768]
    const float* pos    = (const float*)d_in[1]; // [1, 785, 768]
    const float* scores = (const float*)d_in[2]; // [64, 784]
    const float* line   = (const float*)d_in[3]; // [64, 1, 448, 448]
    float* out = (float*)d_out;                  // [64, 392, 768]

    float* partials = (float*)d_ws;                                  // 64*28*3 floats
    int*   sel      = (int*)((char*)d_ws + BATCH * PB * 3 * sizeof(float)); // 64*392 ints

    cog_partial_kernel<<<BATCH * PB, 256, 0, stream>>>(line, partials);
    select_kernel<<<BATCH, 512, 0, stream>>>(partials, scores, sel);
    gather_kernel<<<BATCH * KSEL, 192, 0, stream>>>(magno, pos, sel, out);
}